// SampleOccGrid_72679436583136
// MI455X (gfx1250) — compile-verified
//
#include <hip/hip_runtime.h>
#include <math.h>

#define XDIM 96
#define VOX (96*96*96)       /* 884736 */
#define GDIM 192
#define GVOX (192*192*192)   /* 7077888 */

typedef __attribute__((ext_vector_type(2))) float v2f;
typedef __attribute__((ext_vector_type(8))) float v8f;

// ---------------- zero the bit-packed occupancy grid ----------------
__global__ void zero_u32(unsigned int* p, int n) {
    int i = blockIdx.x * blockDim.x + threadIdx.x;
    if (i < n) p[i] = 0u;
}

// ---------------- per-channel min/max partial reduction ----------------
__global__ __launch_bounds__(256) void reduce_minmax(const float* __restrict__ coords,
                                                     float* __restrict__ pmin,
                                                     float* __restrict__ pmax) {
    __shared__ float smin[256], smax[256];
    int ch = blockIdx.y;                         // b*3 + c
    const float* base = coords + (size_t)ch * VOX;
    int tid = blockIdx.x * blockDim.x + threadIdx.x;   // 0 .. 32767
    float mn = INFINITY, mx = -INFINITY;
    for (int i = tid; i < VOX; i += 128 * 256) {
        float v = base[i];
        mn = fminf(mn, v);
        mx = fmaxf(mx, v);
    }
    smin[threadIdx.x] = mn; smax[threadIdx.x] = mx;
    __syncthreads();
    for (int s = 128; s > 0; s >>= 1) {
        if (threadIdx.x < s) {
            smin[threadIdx.x] = fminf(smin[threadIdx.x], smin[threadIdx.x + s]);
            smax[threadIdx.x] = fmaxf(smax[threadIdx.x], smax[threadIdx.x + s]);
        }
        __syncthreads();
    }
    if (threadIdx.x == 0) {
        pmin[ch * 128 + blockIdx.x] = smin[0];
        pmax[ch * 128 + blockIdx.x] = smax[0];
    }
}

// ---------------- finish reduction + compute transform params ----------------
// params layout (floats): for b<B at b*16: [0..2]=min_voxel_idx, [3..5]=pos_base,
// [6..8]=voxel_size ; at B*16: size_vox[3]
__global__ void params_kernel(const float* __restrict__ pmin, const float* __restrict__ pmax,
                              const float* __restrict__ T, const float* __restrict__ Tinv,
                              float* __restrict__ params, int B) {
    __shared__ float minv[24], maxv[24];
    int t = threadIdx.x;
    int nch = 3 * B;
    if (t < nch) {
        float mn = INFINITY, mx = -INFINITY;
        for (int i = 0; i < 128; ++i) {
            mn = fminf(mn, pmin[t * 128 + i]);
            mx = fmaxf(mx, pmax[t * 128 + i]);
        }
        minv[t] = mn; maxv[t] = mx;
    }
    __syncthreads();
    if (t == 0) {
        // max_size_grid = max_b(max_loc + 0.08 - min_loc)
        float msg[3];
        for (int c = 0; c < 3; ++c) {
            float m = -INFINITY;
            for (int b = 0; b < B; ++b)
                m = fmaxf(m, (maxv[b * 3 + c] + 0.08f) - minv[b * 3 + c]);
            msg[c] = m;
        }
        // size_vox = ceil(max_b(Tinv[:3,:3] @ msg))
        float sv[3];
        for (int i = 0; i < 3; ++i) {
            float m = -INFINITY;
            for (int b = 0; b < B; ++b) {
                const float* Ti = Tinv + b * 16;
                float s = Ti[i * 4 + 0] * msg[0] + Ti[i * 4 + 1] * msg[1] + Ti[i * 4 + 2] * msg[2];
                m = fmaxf(m, s);
            }
            sv[i] = ceilf(m);
            params[B * 16 + i] = sv[i];
        }
        for (int b = 0; b < B; ++b) {
            const float* Ti = Tinv + b * 16;
            const float* Tb = T + b * 16;
            float mvi[3];
            for (int i = 0; i < 3; ++i) {
                float s = Ti[i * 4 + 0] * minv[b * 3 + 0] + Ti[i * 4 + 1] * minv[b * 3 + 1] +
                          Ti[i * 4 + 2] * minv[b * 3 + 2] + Ti[i * 4 + 3];
                mvi[i] = fmaxf(floorf(s), 0.0f);
            }
            float* pp = params + b * 16;
            for (int i = 0; i < 3; ++i) {
                float pb = Tb[i * 4 + 0] * mvi[0] + Tb[i * 4 + 1] * mvi[1] +
                           Tb[i * 4 + 2] * mvi[2] + Tb[i * 4 + 3];
                float ext = Tb[i * 4 + 0] * sv[0] + Tb[i * 4 + 1] * sv[1] + Tb[i * 4 + 2] * sv[2];
                pp[i]     = mvi[i];
                pp[3 + i] = pb;
                pp[6 + i] = ext / sv[i];
            }
        }
    }
}

// ---------------- scatter sparse points into bit-packed occupancy ----------------
__global__ void scatter_kernel(const int* __restrict__ sparse, const float* __restrict__ params,
                               unsigned int* __restrict__ occ, int N, int B, int total) {
    int idx = blockIdx.x * blockDim.x + threadIdx.x;
    if (idx >= total) return;
    int b = idx / N;
    const int* s = sparse + (size_t)idx * 3;
    const float* pp = params + b * 16;
    const float* sv = params + B * 16;
    bool valid = true;
    int ii[3];
#pragma unroll
    for (int i = 0; i < 3; ++i) {
        float se = (float)s[i] - pp[i];
        valid = valid && (se >= 0.0f) && (se < sv[i]);
        int v = (int)se;                       // trunc-toward-zero, matches astype(int32)
        v = v < 0 ? 0 : (v > GDIM - 1 ? GDIM - 1 : v);
        ii[i] = v;
    }
    if (valid) {
        unsigned lin = (((unsigned)(b * GDIM + ii[0]) * GDIM + ii[1]) * GDIM) + ii[2];
        atomicOr(&occ[lin >> 5], 1u << (lin & 31));
    }
}

// ---------------- trilinear sample; emit occ_grids(bool) + byte grid ----------------
__device__ __forceinline__ int clampi(int v, int lo, int hi) {
    return v < lo ? lo : (v > hi ? hi : v);
}

__global__ __launch_bounds__(256) void trilerp_kernel(const float* __restrict__ coords,
                                                      const float* __restrict__ params,
                                                      const unsigned int* __restrict__ occ,
                                                      float* __restrict__ out_occ,
                                                      unsigned char* __restrict__ occb,
                                                      int total) {
    int idx = blockIdx.x * blockDim.x + threadIdx.x;
    if (idx >= total) return;
    int b = idx / VOX;
    int v = idx - b * VOX;
    const float* pp = params + b * 16;
    float f[3];
    int ip[3];
#pragma unroll
    for (int i = 0; i < 3; ++i) {
        float c  = coords[(size_t)(b * 3 + i) * VOX + v];
        float pi = (c - pp[3 + i]) / pp[6 + i] - 0.5f;
        float p0 = floorf(pi);
        f[i]  = pi - p0;
        ip[i] = (int)p0;
    }
    float sum = 0.0f;
#pragma unroll
    for (int dx = 0; dx < 2; ++dx) {
        int ix = clampi(ip[0] + dx, 0, GDIM - 1);
        float wx = dx ? f[0] : 1.0f - f[0];
#pragma unroll
        for (int dy = 0; dy < 2; ++dy) {
            int iy = clampi(ip[1] + dy, 0, GDIM - 1);
            float wxy = wx * (dy ? f[1] : 1.0f - f[1]);
#pragma unroll
            for (int dz = 0; dz < 2; ++dz) {
                int iz = clampi(ip[2] + dz, 0, GDIM - 1);
                float w = wxy * (dz ? f[2] : 1.0f - f[2]);
                unsigned lin = (((unsigned)(b * GDIM + ix) * GDIM + iy) * GDIM) + iz;
                float o = (float)((occ[lin >> 5] >> (lin & 31)) & 1u);
                sum += w * o;
            }
        }
    }
    bool on = sum > 0.0f;          // sum of non-negative terms: >0 iff any term >0
    out_occ[idx] = on ? 1.0f : 0.0f;
    occb[idx]    = on ? 1 : 0;
}

// ---------------- x-axis dilation as banded-ones matmul via WMMA f32 16x16x4 ----
// D(96 x 9216) = A_band(96x96) * M(96x9216) per batch; D>0 == 5-wide box dilate over x.
__global__ __launch_bounds__(32) void xdil_wmma(const unsigned char* __restrict__ in,
                                                unsigned char* __restrict__ out) {
    int lane = threadIdx.x;            // one full wave32 -> EXEC all ones
    int half = lane >> 4;              // 0 | 1
    int l    = lane & 15;
    int n    = blockIdx.x * 16 + l;    // yz column 0..9215
    int i0   = blockIdx.y * 16;        // x row tile
    const unsigned char* base = in + (size_t)blockIdx.z * VOX;
    unsigned char* obase      = out + (size_t)blockIdx.z * VOX;

    v8f acc = {};
    int r = i0 + l;                    // A rows: lanes 0-15 and 16-31 both hold M=0..15
    for (int k0 = 0; k0 < XDIM; k0 += 4) {
        int ka = k0 + 2 * half;        // A/B K layout: VGPR0 -> K=0(lo)/K=2(hi), VGPR1 -> K=1/K=3
        int d0 = r - ka;      if (d0 < 0) d0 = -d0;
        int d1 = r - (ka + 1); if (d1 < 0) d1 = -d1;
        v2f a, bm;
        a.x = (d0 <= 2) ? 1.0f : 0.0f;
        a.y = (d1 <= 2) ? 1.0f : 0.0f;
        bm.x = (float)base[(size_t)ka * 9216 + n];
        bm.y = (float)base[(size_t)(ka + 1) * 9216 + n];
        acc = __builtin_amdgcn_wmma_f32_16x16x4_f32(false, a, false, bm,
                                                    (short)0, acc, false, false);
    }
    // D layout: VGPR j -> row (j + 8*half), col = lane&15
#pragma unroll
    for (int j = 0; j < 8; ++j) {
        obase[(size_t)(i0 + j + 8 * half) * 9216 + n] = (acc[j] > 0.0f) ? 1 : 0;
    }
}

// ---------------- y-axis byte dilation ----------------
__global__ __launch_bounds__(256) void ydil_kernel(const unsigned char* __restrict__ in,
                                                   unsigned char* __restrict__ out, int total) {
    int idx = blockIdx.x * blockDim.x + threadIdx.x;
    if (idx >= total) return;
    int y = (idx / XDIM) % XDIM;
    int lo = y >= 2 ? -2 : -y;
    int hi = y <= XDIM - 3 ? 2 : (XDIM - 1 - y);
    unsigned char m = 0;
    for (int d = lo; d <= hi; ++d) m |= in[idx + d * XDIM];
    out[idx] = m;
}

// ---------------- z-axis byte dilation + mask write (float) ----------------
__global__ __launch_bounds__(256) void zdil_mask_kernel(const unsigned char* __restrict__ in,
                                                        float* __restrict__ mask, int total) {
    int idx = blockIdx.x * blockDim.x + threadIdx.x;
    if (idx >= total) return;
    int z = idx % XDIM;
    int lo = z >= 2 ? -2 : -z;
    int hi = z <= XDIM - 3 ? 2 : (XDIM - 1 - z);
    unsigned char m = 0;
    for (int d = lo; d <= hi; ++d) m |= in[idx + d];
    mask[idx] = m ? 1.0f : 0.0f;
}

// ---------------- scatter rand_idx Trues into mask ----------------
__global__ void rand_kernel(const int* __restrict__ ridx, float* __restrict__ mask,
                            int num, int B) {
    int i = blockIdx.x * blockDim.x + threadIdx.x;
    if (i >= num) return;
    int b = i % B;
    int x = ridx[i], y = ridx[num + i], z = ridx[2 * num + i];
    mask[(size_t)b * VOX + (size_t)x * (XDIM * XDIM) + y * XDIM + z] = 1.0f;
}

extern "C" void kernel_launch(void* const* d_in, const int* in_sizes, int n_in,
                              void* d_out, int out_size, void* d_ws, size_t ws_size,
                              hipStream_t stream) {
    const float* coords = (const float*)d_in[0];
    const float* T      = (const float*)d_in[1];
    const float* Tinv   = (const float*)d_in[2];
    const int*   sparse = (const int*)d_in[3];
    const int*   ridx   = (const int*)d_in[5];

    int B      = in_sizes[0] / (3 * VOX);
    int N      = in_sizes[3] / (3 * B);
    int numAdd = in_sizes[5] / 3;

    // workspace carve-up
    char* ws = (char*)d_ws;
    size_t off = 0;
    float* pmin = (float*)(ws + off); off += (size_t)B * 3 * 128 * sizeof(float);
    float* pmax = (float*)(ws + off); off += (size_t)B * 3 * 128 * sizeof(float);
    float* params = (float*)(ws + off); off += 256 * sizeof(float);
    off = (off + 255) & ~(size_t)255;
    size_t nwords = (size_t)B * GVOX / 32;
    unsigned int* occbits = (unsigned int*)(ws + off); off += nwords * 4;
    unsigned char* occb = (unsigned char*)(ws + off); off += (size_t)B * VOX;
    unsigned char* tmpA = (unsigned char*)(ws + off); off += (size_t)B * VOX;

    float* out_occ  = (float*)d_out;
    float* out_mask = out_occ + (size_t)B * VOX;

    int totalVox = B * VOX;
    int totalPts = B * N;

    zero_u32<<<(int)((nwords + 255) / 256), 256, 0, stream>>>(occbits, (int)nwords);
    reduce_minmax<<<dim3(128, B * 3), 256, 0, stream>>>(coords, pmin, pmax);
    params_kernel<<<1, 64, 0, stream>>>(pmin, pmax, T, Tinv, params, B);
    scatter_kernel<<<(totalPts + 255) / 256, 256, 0, stream>>>(sparse, params, occbits, N, B, totalPts);
    trilerp_kernel<<<(totalVox + 255) / 256, 256, 0, stream>>>(coords, params, occbits,
                                                               out_occ, occb, totalVox);
    // x-dilation via WMMA banded matmul: grid (9216/16, 96/16, B), one wave per tile
    xdil_wmma<<<dim3(9216 / 16, XDIM / 16, B), 32, 0, stream>>>(occb, tmpA);
    ydil_kernel<<<(totalVox + 255) / 256, 256, 0, stream>>>(tmpA, occb, totalVox);
    zdil_mask_kernel<<<(totalVox + 255) / 256, 256, 0, stream>>>(occb, out_mask, totalVox);
    rand_kernel<<<(numAdd + 255) / 256, 256, 0, stream>>>(ridx, out_mask, numAdd, B);
}